// XSimGCL_Encoder_33217277067601
// MI455X (gfx1250) — compile-verified
//
#include <hip/hip_runtime.h>
#include <hip/hip_bf16.h>
#include <stdint.h>

#define USER_NUM 100000
#define ITEM_NUM 50000
#define NTOT     (USER_NUM + ITEM_NUM)
#define EMB      64
#define NNZ_     1200000
#define N_LAYERS 3
#define EPS_F    0.2f
#define TILE     256   // edges staged per block-iteration via async global->LDS

// ----------------------------------------------------------------------------
// ego = concat(user_emb, item_emb), float4 copies
// ----------------------------------------------------------------------------
__global__ void init_ego(const float* __restrict__ ue,
                         const float* __restrict__ ie,
                         float* __restrict__ ego) {
    size_t i = (size_t)blockIdx.x * blockDim.x + threadIdx.x;   // float4 units
    const size_t total4 = (size_t)NTOT * EMB / 4;
    const size_t user4  = (size_t)USER_NUM * EMB / 4;
    if (i >= total4) return;
    float4 v = (i < user4) ? ((const float4*)ue)[i]
                           : ((const float4*)ie)[i - user4];
    ((float4*)ego)[i] = v;
}

// ----------------------------------------------------------------------------
// zero a N*EMB float buffer, float4 stores
// ----------------------------------------------------------------------------
__global__ void zero_buf(float* __restrict__ p) {
    size_t i = (size_t)blockIdx.x * blockDim.x + threadIdx.x;
    const size_t total4 = (size_t)NTOT * EMB / 4;
    if (i >= total4) return;
    ((float4*)p)[i] = make_float4(0.f, 0.f, 0.f, 0.f);
}

// ----------------------------------------------------------------------------
// Edge SpMM: for each edge e: ego_out[rows[e]] += ego_in[cols[e]] * vals[e]
//
// Persistent blocks, double-buffered async pipeline:
//   - waves 0..5 stage the next tile's rows/cols/vals into the alternate LDS
//     buffer with global_load_async_to_lds_b128 (1 op/thread, 4 edges each)
//   - s_wait_asynccnt 1 overlaps staging of tile i+1 with waiting on tile i
//     (async loads complete in order per wave)
//   - consumers: 16-lane edge groups, b128 gather + 4 f32 L2 atomics/lane.
// Both ego buffers (38.4 MB each) are L2-resident on the 192 MB L2.
// ----------------------------------------------------------------------------
__global__ void edge_spmm(const float* __restrict__ ego_in,
                          float* __restrict__ ego_out,
                          const int* __restrict__ rows,
                          const int* __restrict__ cols,
                          const float* __restrict__ vals,
                          int nnz) {
    __shared__ int   s_rows[2][TILE];
    __shared__ int   s_cols[2][TILE];
    __shared__ float s_vals[2][TILE];

    const int t      = threadIdx.x;
    const int ntiles = (nnz + TILE - 1) / TILE;

    // Stage one tile (4 edges per thread, b128). Role is uniform per wave
    // (64-thread role groups = 2 whole wave32s), so the global base pointer
    // legitimately lives in an SGPR pair.
    auto stage = [&](int buf, int tile_idx) {
        if (t < 192) {
            const int role = t >> 6;              // 0=rows 1=cols 2=vals
            const int j    = (t & 63) << 2;       // edge slot within tile
            int src = tile_idx * TILE + j;
            if (src > nnz - 4) src = nnz - 4;     // keep b128 in bounds
            uint32_t voff = (uint32_t)src * 4u;   // byte offset
            const void* gbase;
            void*       laddr;
            if (role == 0)      { gbase = (const void*)rows; laddr = (void*)&s_rows[buf][j]; }
            else if (role == 1) { gbase = (const void*)cols; laddr = (void*)&s_cols[buf][j]; }
            else                { gbase = (const void*)vals; laddr = (void*)&s_vals[buf][j]; }
            // generic pointer to __shared__ carries the LDS byte offset in
            // its low 32 bits — exactly what the async VDST operand needs.
            uint32_t lds = (uint32_t)(uintptr_t)laddr;
            asm volatile("global_load_async_to_lds_b128 %0, %1, %2"
                         :: "v"(lds), "v"(voff), "s"(gbase) : "memory");
        }
    };

    int tile = blockIdx.x;
    if (tile >= ntiles) return;

    const int g = t >> 4;          // 16 groups of 16 lanes
    const int c = (t & 15) * 4;    // float column within the 64-wide row

    int cur = 0;
    stage(cur, tile);

    while (true) {
        const int  next      = tile + gridDim.x;
        const bool have_next = next < ntiles;
        if (have_next) {
            stage(cur ^ 1, next);
            asm volatile("s_wait_asynccnt 1" ::: "memory");  // tile `cur` landed
        } else {
            asm volatile("s_wait_asynccnt 0" ::: "memory");
        }
        __syncthreads();

        const int base = tile * TILE;
        #pragma unroll 4
        for (int i = 0; i < 16; ++i) {
            int el = g * 16 + i;                 // consecutive within group
            int eg = base + el;
            if (eg >= nnz) break;
            int   col = s_cols[cur][el];         // LDS broadcast in group
            int   row = s_rows[cur][el];
            float val = s_vals[cur][el];
            float4 v = *(const float4*)(ego_in + (size_t)col * EMB + c);
            float* dst = ego_out + (size_t)row * EMB + c;
            atomicAdd(dst + 0, v.x * val);
            atomicAdd(dst + 1, v.y * val);
            atomicAdd(dst + 2, v.z * val);
            atomicAdd(dst + 3, v.w * val);
        }
        __syncthreads();            // all waves done reading `cur` before it
                                    // gets restaged next iteration
        if (!have_next) break;
        cur ^= 1;
        tile = next;
    }
}

// ----------------------------------------------------------------------------
// Layer epilogue: one wave32 per node.
//   ego += sign(ego) * (noise / max(||noise||,1e-12)) * EPS   (in place)
//   acc (+)= ego / 3 ;  cl = ego at layer LAYER_CL-1
// Norm via 32-lane shfl_xor reduction over the 64-wide row (float2/lane).
// ----------------------------------------------------------------------------
__global__ void post_layer(float* __restrict__ ego,
                           const float* __restrict__ noise,   // layer slice
                           float* __restrict__ acc,           // d_out[0..N*EMB)
                           float* __restrict__ cl,            // d_out[N*EMB..)
                           int write_cl, int first_layer) {
    const int waves_per_block = blockDim.x >> 5;               // wave32
    int node = blockIdx.x * waves_per_block + (threadIdx.x >> 5);
    int lane = threadIdx.x & 31;
    if (node >= NTOT) return;

    size_t rb = (size_t)node * EMB + (size_t)lane * 2;
    float2 nz = *(const float2*)(noise + rb);

    float ss = nz.x * nz.x + nz.y * nz.y;
    #pragma unroll
    for (int m = 16; m >= 1; m >>= 1) ss += __shfl_xor(ss, m, 32);
    float scale = EPS_F / fmaxf(sqrtf(ss), 1e-12f);

    float2 ev = *(float2*)(ego + rb);
    float sx = (ev.x > 0.f) ? 1.f : ((ev.x < 0.f) ? -1.f : 0.f);
    float sy = (ev.y > 0.f) ? 1.f : ((ev.y < 0.f) ? -1.f : 0.f);
    ev.x += sx * nz.x * scale;
    ev.y += sy * nz.y * scale;
    *(float2*)(ego + rb) = ev;                 // becomes next layer's input

    const float third = 1.0f / (float)N_LAYERS;
    float2 a;
    if (first_layer) { a.x = ev.x * third;            a.y = ev.y * third; }
    else             { a = *(float2*)(acc + rb);
                       a.x += ev.x * third;           a.y += ev.y * third; }
    *(float2*)(acc + rb) = a;

    if (write_cl) *(float2*)(cl + rb) = ev;
}

// ----------------------------------------------------------------------------
extern "C" void kernel_launch(void* const* d_in, const int* in_sizes, int n_in,
                              void* d_out, int out_size, void* d_ws, size_t ws_size,
                              hipStream_t stream) {
    (void)in_sizes; (void)n_in; (void)out_size; (void)ws_size;

    const float* ue    = (const float*)d_in[0];
    const float* ie    = (const float*)d_in[1];
    const int*   rows  = (const int*)  d_in[2];
    const int*   cols  = (const int*)  d_in[3];
    const float* vals  = (const float*)d_in[4];
    const float* noise = (const float*)d_in[5];

    float* out_final = (float*)d_out;                       // (u,i) = final[0:N]
    float* out_cl    = out_final + (size_t)NTOT * EMB;      // (u_cl,i_cl)

    float* ego_a = (float*)d_ws;                            // ping
    float* ego_b = ego_a + (size_t)NTOT * EMB;              // pong

    const int tb = 256;
    const size_t total4 = (size_t)NTOT * EMB / 4;
    const int cp_blocks = (int)((total4 + tb - 1) / tb);

    init_ego<<<cp_blocks, tb, 0, stream>>>(ue, ie, ego_a);

    const int ntiles = (NNZ_ + TILE - 1) / TILE;            // 4688
    int edge_blocks = 1172;                                 // 4 tiles/block
    if (edge_blocks > ntiles) edge_blocks = ntiles;
    const int waves_per_block = tb / 32;
    const int post_blocks = (NTOT + waves_per_block - 1) / waves_per_block;

    for (int k = 0; k < N_LAYERS; ++k) {
        zero_buf<<<cp_blocks, tb, 0, stream>>>(ego_b);
        edge_spmm<<<edge_blocks, tb, 0, stream>>>(ego_a, ego_b, rows, cols, vals, NNZ_);
        post_layer<<<post_blocks, tb, 0, stream>>>(
            ego_b, noise + (size_t)k * NTOT * EMB,
            out_final, out_cl,
            /*write_cl=*/(k == 0) ? 1 : 0,       // LAYER_CL - 1 == 0
            /*first_layer=*/(k == 0) ? 1 : 0);
        float* tmp = ego_a; ego_a = ego_b; ego_b = tmp;
    }
}